// DiverseBeamSearch_72885595013691
// MI455X (gfx1250) — compile-verified
//
#include <hip/hip_runtime.h>
#include <hip/hip_bf16.h>
#include <math.h>

#define GQ 4             // NUM_GROUPS
#define NTHREADS 1024    // 32 waves (CDNA5 wave32)
#define NWAVES 32
#define DEPTH 8          // ring slots per wave
#define SLOT_FLOATS 256  // 1KB slot = 2 x b128 async ops (8 floats per lane)
#define SLOT_BYTES 1024
#define MAXPEN 6         // <= 2*(G-1) penalized tokens
#define NROWSTREAM (2 * GQ)   // 8 row-streams per batch, group-major order

// ---- ordering identical to jax.lax.top_k: value desc, index asc on ties ----
__device__ __forceinline__ bool better(float av, int ai, float bv, int bi) {
    return (av > bv) || ((av == bv) && (ai < bi));
}
__device__ __forceinline__ void upd2(float v, int i,
                                     float& t0v, int& t0i, float& t1v, int& t1i) {
    if (better(v, i, t0v, t0i)) { t1v = t0v; t1i = t0i; t0v = v; t0i = i; }
    else if (better(v, i, t1v, t1i)) { t1v = v; t1i = i; }
}

// Low 32 bits of a generic __shared__ pointer = raw LDS byte offset (ISA 10.2).
__device__ __forceinline__ unsigned lds_lo32(const void* p) {
    return (unsigned)(unsigned long long)p;
}
// CDNA5 async DMA: lane copies 16B from global address sbase plus voff to LDS offset lds_byte; ASYNCcnt.
__device__ __forceinline__ void async_copy_b128(unsigned lds_byte, unsigned voff,
                                                unsigned long long sbase) {
    asm volatile("global_load_async_to_lds_b128 %0, %1, %2"
                 :: "v"(lds_byte), "v"(voff), "s"(sbase)
                 : "memory");
}
// 2 async instrs per slot, DEPTH slots: full = 16 outstanding; <=14 => oldest slot done.
__device__ __forceinline__ void wait_async_steady() {
    asm volatile("s_wait_asynccnt 14" ::: "memory");
}
__device__ __forceinline__ void wait_async_0() {
    asm volatile("s_wait_asynccnt 0" ::: "memory");
}

__global__ void dbs_init_go(const float* __restrict__ go, float* __restrict__ out_go, int n) {
    int i = blockIdx.x * blockDim.x + threadIdx.x;
    if (i < n) out_go[i] = 0.5f * go[i];
}

__global__ __launch_bounds__(NTHREADS)
void dbs_main(const float* __restrict__ lprobs,
              const float* __restrict__ scores,
              const float* __restrict__ group_overlap,
              const int* __restrict__ mask,
              const long long* __restrict__ orig,
              const int* __restrict__ step_ptr,
              float* __restrict__ out_scores,
              float* __restrict__ out_idx,
              float* __restrict__ out_beams,
              float* __restrict__ out_go,
              int beam, int vocab, int nstep) {
    // 256KB async stream ring: needs CDNA5 320KB-per-workgroup LDS.
    __shared__ __align__(16) float s_stream[NWAVES][DEPTH][SLOT_FLOATS];
    __shared__ float s_pv0[NWAVES], s_pv1[NWAVES];
    __shared__ int   s_pi0[NWAVES], s_pi1[NWAVES];
    __shared__ float s_gob[GQ * GQ];
    __shared__ int   s_mask[GQ * 2];
    __shared__ float s_score[GQ * 2];
    __shared__ int   s_ptok[MAXPEN + 2];
    __shared__ float s_ppen[MAXPEN + 2];
    __shared__ int   s_chosen[GQ][2];

    const int b = blockIdx.x, t = threadIdx.x, w = t >> 5, lane = t & 31;
    const long long ob = orig[b];
    const int step = step_ptr[0];

    if (t < GQ * GQ) s_gob[t] = group_overlap[(size_t)ob * (GQ * GQ) + t];
    if (t < beam) {
        s_mask[t]  = mask[b * beam + t];
        s_score[t] = scores[((size_t)(b * beam + t)) * nstep + (step - 1)];
    }
    __syncthreads();

    const float* batchptr = lprobs + (size_t)b * beam * (size_t)vocab;
    const unsigned long long batchbase = (unsigned long long)batchptr;
    const int Cfull = vocab / SLOT_FLOATS;               // full 1KB slots per row
    const int TAIL  = vocab - Cfull * SLOT_FLOATS;
    const unsigned lane16  = (unsigned)lane * 16u;
    const unsigned ldsbase = lds_lo32(&s_stream[w][0][0]) + lane16;

    // Per-wave cursors over the global slot sequence (row-stream m = 0..7,
    // row(m) = (m&1)*4 + (m>>1): group-major so issues run ahead of barriers).
    int iss_m = 0, iss_cc = w;
    while (iss_cc >= Cfull) { iss_cc -= Cfull; ++iss_m; }
    int proc_m = iss_m, proc_cc = iss_cc;
    int kslot = 0;

    // ---- prologue: fill DEPTH-deep ring; issues are penalty-independent ----
    #pragma unroll
    for (int d = 0; d < DEPTH; ++d) {
        if (iss_m < NROWSTREAM) {
            const int row = ((iss_m & 1) << 2) + (iss_m >> 1);
            const unsigned voff = (unsigned)(row * vocab + iss_cc * SLOT_FLOATS) * 4u + lane16;
            const unsigned ldst = ldsbase + (unsigned)(d * SLOT_BYTES);
            async_copy_b128(ldst,        voff,        batchbase);
            async_copy_b128(ldst + 512u, voff + 512u, batchbase);
            iss_cc += NWAVES;
            while (iss_cc >= Cfull) { iss_cc -= Cfull; ++iss_m; }
        }
    }

    for (int g = 0; g < GQ; ++g) {
        // --- penalty table from previously chosen tokens ---
        if (t < 2 * g) {
            const int gp = t >> 1, j = t & 1;
            s_ptok[t] = s_chosen[gp][j];
            s_ppen[t] = -0.5f * (1.0f + s_gob[g * GQ + gp]); // DIVERSITY_STRENGTH*(1+go)
        }
        __syncthreads();                                    // A: pen table ready
        const int np = 2 * g;
        int rtok[MAXPEN]; float rpen[MAXPEN];
        #pragma unroll
        for (int i = 0; i < MAXPEN; ++i) {
            rtok[i] = (i < np) ? s_ptok[i] : -1;            // -1 never matches
            rpen[i] = (i < np) ? s_ppen[i] : 0.0f;
        }
        const float rsc0 = s_score[g], rsc1 = s_score[GQ + g];
        const int   mr0  = s_mask[g],  mr1  = s_mask[GQ + g];

        // tail prefetch now; latency hidden behind the slot loop
        float tailv = 0.0f; int tail_sr = 0, tail_v = 0, have_tail = 0;
        if (t < 2 * TAIL) {
            tail_sr = (t >= TAIL);
            tail_v  = Cfull * SLOT_FLOATS + (t - (tail_sr ? TAIL : 0));
            tailv   = batchptr[(size_t)(tail_sr * GQ + g) * vocab + tail_v];
            have_tail = 1;
        }

        float t0v = -INFINITY, t1v = -INFINITY;
        int   t0i = 0x7fffffff, t1i = 0x7fffffff;

        const int mEnd = 2 * (g + 1);
        while (proc_m < mEnd) {
            const int   sr      = proc_m & 1;
            const float rsc     = sr ? rsc1 : rsc0;
            const int   mrow    = sr ? mr1  : mr0;
            const int   rowbase = sr ? vocab : 0;

            if (iss_m < NROWSTREAM) wait_async_steady();    // oldest slot landed
            else                    wait_async_0();         // final drain (once)

            const float* buf = &s_stream[w][kslot][0];
            const float4 da = *(const float4*)(buf + lane * 4);
            const float4 db = *(const float4*)(buf + 128 + lane * 4);
            // penalties are strictly negative: unpenalized slot-max is an upper bound
            const float cmax = fmaxf(fmaxf(fmaxf(da.x, da.y), fmaxf(da.z, da.w)),
                                     fmaxf(fmaxf(db.x, db.y), fmaxf(db.z, db.w)));
            const float adj = mrow ? (cmax + rsc) : rsc;
            if (adj > t1v) {                                // rare path: exact update
                const int tb = proc_cc * SLOT_FLOATS + lane * 4;
                const float va[8] = {da.x, da.y, da.z, da.w, db.x, db.y, db.z, db.w};
                #pragma unroll
                for (int j2 = 0; j2 < 8; ++j2) {
                    const int tok = tb + ((j2 < 4) ? j2 : (124 + j2)); // +128 block
                    float val;
                    if (mrow) {
                        val = va[j2];
                        #pragma unroll
                        for (int i = 0; i < MAXPEN; ++i)
                            if (tok == rtok[i]) val += rpen[i];
                        val += rsc;
                    } else val = rsc;                       // masked row => score only
                    upd2(val, rowbase + tok, t0v, t0i, t1v, t1i);
                }
            }
            // refill the just-freed slot; may belong to a LATER group (no dependency)
            if (iss_m < NROWSTREAM) {
                const int row = ((iss_m & 1) << 2) + (iss_m >> 1);
                const unsigned voff = (unsigned)(row * vocab + iss_cc * SLOT_FLOATS) * 4u + lane16;
                const unsigned ldst = ldsbase + (unsigned)(kslot * SLOT_BYTES);
                async_copy_b128(ldst,        voff,        batchbase);
                async_copy_b128(ldst + 512u, voff + 512u, batchbase);
                iss_cc += NWAVES;
                while (iss_cc >= Cfull) { iss_cc -= Cfull; ++iss_m; }
            }
            kslot = (kslot + 1 == DEPTH) ? 0 : kslot + 1;
            proc_cc += NWAVES;
            while (proc_cc >= Cfull) { proc_cc -= Cfull; ++proc_m; }
        }

        // tail consume
        if (have_tail) {
            const float rsc  = tail_sr ? rsc1 : rsc0;
            const int   mrow = tail_sr ? mr1  : mr0;
            float val;
            if (mrow) {
                val = tailv;
                #pragma unroll
                for (int i = 0; i < MAXPEN; ++i)
                    if (tail_v == rtok[i]) val += rpen[i];
                val += rsc;
            } else val = rsc;
            upd2(val, tail_sr * vocab + tail_v, t0v, t0i, t1v, t1i);
        }

        // ---- intra-wave top-2 xor-reduction (5 shfl steps, no barriers) ----
        #pragma unroll
        for (int m = 16; m >= 1; m >>= 1) {
            const float ov0 = __shfl_xor(t0v, m); const int oi0 = __shfl_xor(t0i, m);
            const float ov1 = __shfl_xor(t1v, m); const int oi1 = __shfl_xor(t1i, m);
            float r0v, r1v; int r0i, r1i;
            if (better(t0v, t0i, ov0, oi0)) {
                r0v = t0v; r0i = t0i;
                if (better(t1v, t1i, ov0, oi0)) { r1v = t1v; r1i = t1i; }
                else                            { r1v = ov0; r1i = oi0; }
            } else {
                r0v = ov0; r0i = oi0;
                if (better(ov1, oi1, t0v, t0i)) { r1v = ov1; r1i = oi1; }
                else                            { r1v = t0v; r1i = t0i; }
            }
            t0v = r0v; t0i = r0i; t1v = r1v; t1i = r1i;
        }
        if (lane == 0) { s_pv0[w] = t0v; s_pi0[w] = t0i; s_pv1[w] = t1v; s_pi1[w] = t1i; }
        __syncthreads();                                    // B
        if (w == 0) {                                       // 32 partials -> wave 0
            float p0v = s_pv0[lane], p1v = s_pv1[lane];
            int   p0i = s_pi0[lane], p1i = s_pi1[lane];
            #pragma unroll
            for (int m = 16; m >= 1; m >>= 1) {
                const float ov0 = __shfl_xor(p0v, m); const int oi0 = __shfl_xor(p0i, m);
                const float ov1 = __shfl_xor(p1v, m); const int oi1 = __shfl_xor(p1i, m);
                float r0v, r1v; int r0i, r1i;
                if (better(p0v, p0i, ov0, oi0)) {
                    r0v = p0v; r0i = p0i;
                    if (better(p1v, p1i, ov0, oi0)) { r1v = p1v; r1i = p1i; }
                    else                            { r1v = ov0; r1i = oi0; }
                } else {
                    r0v = ov0; r0i = oi0;
                    if (better(ov1, oi1, p0v, p0i)) { r1v = ov1; r1i = oi1; }
                    else                            { r1v = p0v; r1i = p0i; }
                }
                p0v = r0v; p0i = r0i; p1v = r1v; p1i = r1i;
            }
            if (lane == 0) {
                const int ss0 = p0i / vocab, tok0 = p0i - ss0 * vocab;
                const int ss1 = p1i / vocab, tok1 = p1i - ss1 * vocab;
                const int o0 = b * (GQ * 2) + g;            // [b, j*G+g] layout
                const int o1 = b * (GQ * 2) + GQ + g;
                out_scores[o0] = p0v;             out_scores[o1] = p1v;
                out_idx[o0]    = (float)tok0;     out_idx[o1]    = (float)tok1;
                out_beams[o0]  = (float)(ss0 * GQ + g);
                out_beams[o1]  = (float)(ss1 * GQ + g);
                s_chosen[g][0] = tok0;            s_chosen[g][1] = tok1;
            }
        }
        __syncthreads();                                    // C: s_chosen visible
    }

    // ---- 4x4 overlap update: new_go = (go + overlap) * 0.5 ----
    if (t < GQ * GQ) {
        const int a = t >> 2, c = t & 3;
        int ov = 0;
        #pragma unroll
        for (int j = 0; j < 2; ++j) {
            if (s_mask[j * GQ + a] && s_mask[j * GQ + c] &&
                (s_chosen[a][j] == s_chosen[c][j])) ov++;
        }
        atomicAdd(&out_go[(size_t)ob * (GQ * GQ) + t], 0.5f * (float)ov);
    }
}

extern "C" void kernel_launch(void* const* d_in, const int* in_sizes, int n_in,
                              void* d_out, int out_size, void* d_ws, size_t ws_size,
                              hipStream_t stream) {
    const float*     lprobs = (const float*)d_in[0];
    const float*     scores = (const float*)d_in[1];
    const float*     go     = (const float*)d_in[2];
    const int*       mask   = (const int*)d_in[3];
    const long long* orig   = (const long long*)d_in[4];
    const int*       step   = (const int*)d_in[5];

    const int bsz   = in_sizes[4];
    const int beam  = in_sizes[3] / bsz;
    const int vocab = in_sizes[0] / (bsz * beam);
    const int nstep = in_sizes[1] / (bsz * beam);

    float* out        = (float*)d_out;
    float* out_scores = out;                          // (bsz, beam)
    float* out_idx    = out + (size_t)bsz * beam;     // (bsz, beam)
    float* out_beams  = out + (size_t)2 * bsz * beam; // (bsz, beam)
    float* out_go     = out + (size_t)3 * bsz * beam; // (bsz, 4, 4)

    const int ngo = bsz * GQ * GQ;
    dbs_init_go<<<(ngo + 255) / 256, 256, 0, stream>>>(go, out_go, ngo);
    dbs_main<<<bsz, NTHREADS, 0, stream>>>(lprobs, scores, go, mask, orig, step,
                                           out_scores, out_idx, out_beams, out_go,
                                           beam, vocab, nstep);
}